// PointerNetwork_22136261444340
// MI455X (gfx1250) — compile-verified
//
#include <hip/hip_runtime.h>
#include <hip/hip_bf16.h>

typedef __attribute__((ext_vector_type(2))) float v2f;
typedef __attribute__((ext_vector_type(8))) float v8f;

constexpr int kB   = 32;
constexpr int kS   = 256;
constexpr int kE   = 512;
constexpr int kH   = 512;
constexpr int kW   = 512;
constexpr int kENC = 1024;
constexpr int k4H  = 2048;

// ---------------------------------------------------------------------------
// Phase 1: bsum[r,w] = enc[r,:]@W1 + emb[wid[r],:]@W2  for r in [0, B*S)
// One wave computes one 16x16 fp32 output tile via V_WMMA_F32_16X16X4_F32.
// Tile grid: 512 row-tiles x 32 col-tiles = 16384 waves; 8 waves/block
// (waves in a block share tile_m -> A loads hit in WGP$).
// ---------------------------------------------------------------------------
__global__ __launch_bounds__(256) void bsum_wmma_kernel(
    const float* __restrict__ enc, const int* __restrict__ wid,
    const float* __restrict__ emb, const float* __restrict__ W1,
    const float* __restrict__ W2, float* __restrict__ bsum)
{
  const int lane = threadIdx.x & 31;
  const int wave = blockIdx.x * 8 + (threadIdx.x >> 5);
  const int tile_n = wave & 31;      // 32 col tiles of 16
  const int tile_m = wave >> 5;      // 512 row tiles of 16
  const int half = lane >> 4;        // 0: K=k,k+1   1: K=k+2,k+3
  const int l16  = lane & 15;

  const int rowA = tile_m * 16 + l16;   // A row held by this lane
  const int colB = tile_n * 16 + l16;   // B/D column held by this lane

  v8f acc = {};

  // K over encoder features (ENC = 1024)
  const float* arow = enc + (size_t)rowA * kENC;
  for (int k = 0; k < kENC; k += 4) {
    v2f a = *(const v2f*)(arow + k + 2 * half);
    v2f bf;
    bf.x = W1[(size_t)(k + 2 * half) * kW + colB];
    bf.y = W1[(size_t)(k + 2 * half + 1) * kW + colB];
    acc = __builtin_amdgcn_wmma_f32_16x16x4_f32(
        false, a, false, bf, (short)0, acc, false, false);
  }

  // K over gathered word embedding (E = 512)
  const float* grow = emb + (size_t)wid[rowA] * kE;
  for (int k = 0; k < kE; k += 4) {
    v2f a = *(const v2f*)(grow + k + 2 * half);
    v2f bf;
    bf.x = W2[(size_t)(k + 2 * half) * kW + colB];
    bf.y = W2[(size_t)(k + 2 * half + 1) * kW + colB];
    acc = __builtin_amdgcn_wmma_f32_16x16x4_f32(
        false, a, false, bf, (short)0, acc, false, false);
  }

  // C/D layout: VGPR v -> row tile_m*16 + v + 8*half, col = colB
  #pragma unroll
  for (int v = 0; v < 8; ++v) {
    int r = tile_m * 16 + v + 8 * half;
    bsum[(size_t)r * kW + colB] = acc[v];
  }
}

// ---------------------------------------------------------------------------
// Phase 2: sequential pointer decode. One workgroup (1024 threads = 32 waves)
// per batch element; 256 dependent steps; state in LDS; weights stream from
// L2 with b128 (float4) loads to maximize per-WGP load throughput.
// ---------------------------------------------------------------------------
__global__ __launch_bounds__(1024) void decode_kernel(
    const int* __restrict__ word_ids, const float* __restrict__ enc_hn,
    const float* __restrict__ emb, const float* __restrict__ chunk_emb,
    const float* __restrict__ W3, const float* __restrict__ W4,
    const float* __restrict__ vt1, const float* __restrict__ vt2,
    const float* __restrict__ Wx, const float* __restrict__ Wh,
    const float* __restrict__ bias, const float* __restrict__ bsum,
    float* __restrict__ out)
{
  __shared__ int    wid[kS];
  __shared__ float  xmean[kE], rowb[kE], hvec[kH], cvec[kH];
  __shared__ float4 gx4[k4H / 4], gh4[k4H / 4];      // gates partials (x@Wx, h@Wh)
  __shared__ float4 u3a4[kW / 4], u4a4[kW / 4];      // attention partials
  __shared__ float4 vt1s4[kW / 4];                   // vt1 cached in LDS
  __shared__ float  biass[k4H];                      // bias cached in LDS
  __shared__ float  scores[kS], outv[kS];
  __shared__ float  rval[kS];
  __shared__ int    ridx[kS];
  __shared__ float  ce_s;
  __shared__ int    s_last, s_bond, s_fin;

  const int b    = blockIdx.x;
  const int tid  = threadIdx.x;
  const int lane = tid & 31;
  const int wv   = tid >> 5;
  const float* bsum_b = bsum + (size_t)b * kS * kW;

  // one-time initialization
  for (int i = tid; i < kS; i += 1024) { wid[i] = word_ids[b * kS + i]; outv[i] = 0.f; }
  for (int i = tid; i < kH; i += 1024) { hvec[i] = 0.f; cvec[i] = enc_hn[b * kH + i]; }
  for (int i = tid; i < k4H; i += 1024) biass[i] = bias[i];
  if (tid < kW / 4) vt1s4[tid] = *(const float4*)(vt1 + 4 * tid);
  if (tid == 0) { s_last = 0; s_bond = 0; s_fin = 0; }
  __syncthreads();

  const float* gxv = (const float*)gx4;
  const float* ghv = (const float*)gh4;
  const float* u3v = (const float*)u3a4;
  const float* u4v = (const float*)u4a4;

  for (int step = 0; step < kS; ++step) {
    const int last_bond = s_last;
    const int bond      = s_bond;   // bond < kS here, so bidx == bond

    // A) segment mean (threads 0..511) || emb row at bidx (threads 512..1023)
    if (tid < kE) {
      float sum = 0.f;
      for (int s = last_bond; s < bond; ++s) sum += emb[(size_t)wid[s] * kE + tid];
      int cnt = bond - last_bond;
      xmean[tid] = sum / (float)(cnt > 0 ? cnt : 1);
    } else {
      int e = tid - kE;
      rowb[e] = emb[(size_t)wid[bond] * kE + e];
    }
    __syncthreads();

    // B) gates partials: mat0 = x@Wx (tid<512), mat1 = h@Wh (tid>=512)
    {
      const int t  = tid & 511;
      const int cb = 4 * t;                       // 4-column group
      const float* wp  = (tid < 512) ? (Wx + cb) : (Wh + cb);
      const float* xv  = (tid < 512) ? xmean : hvec;
      float a0 = 0.f, a1 = 0.f, a2 = 0.f, a3 = 0.f;
      for (int e = 0; e < kE; ++e) {
        float4 w4 = *(const float4*)(wp + (size_t)e * k4H);
        float  x  = xv[e];
        a0 += x * w4.x; a1 += x * w4.y; a2 += x * w4.z; a3 += x * w4.w;
      }
      float4 r; r.x = a0; r.y = a1; r.z = a2; r.w = a3;
      if (tid < 512) gx4[t] = r; else gh4[t] = r;
    }
    __syncthreads();

    // C) LSTM cell update (threads 0..511)
    if (tid < kH) {
      int j = tid;
      float gi = biass[j]          + gxv[j]          + ghv[j];
      float gf = biass[kH + j]     + gxv[kH + j]     + ghv[kH + j];
      float gg = biass[2 * kH + j] + gxv[2 * kH + j] + ghv[2 * kH + j];
      float go = biass[3 * kH + j] + gxv[3 * kH + j] + ghv[3 * kH + j];
      float ig = 1.f / (1.f + __expf(-gi));
      float fg = 1.f / (1.f + __expf(-gf));
      float gt = tanhf(gg);
      float og = 1.f / (1.f + __expf(-go));
      float cn = fg * cvec[j] + ig * gt;
      cvec[j] = cn;
      hvec[j] = og * tanhf(cn);
    }
    __syncthreads();

    // D) u3 = inp[bidx]@W3 (tid<256), u4 = h@W4 (tid 256..511), ce (wave 16)
    if (tid < 256) {
      const int cb = 4 * tid;
      float a0 = 0.f, a1 = 0.f, a2 = 0.f, a3 = 0.f;
      for (int e = 0; e < kE; ++e) {
        float4 w4 = *(const float4*)(W3 + (size_t)e * kW + cb);
        float  x  = rowb[e];
        a0 += x * w4.x; a1 += x * w4.y; a2 += x * w4.z; a3 += x * w4.w;
      }
      float4 r; r.x = a0; r.y = a1; r.z = a2; r.w = a3;
      u3a4[tid] = r;
    } else if (tid < 512) {
      const int g  = tid - 256;
      const int cb = 4 * g;
      float a0 = 0.f, a1 = 0.f, a2 = 0.f, a3 = 0.f;
      for (int e = 0; e < kH; ++e) {
        float4 w4 = *(const float4*)(W4 + (size_t)e * kW + cb);
        float  x  = hvec[e];
        a0 += x * w4.x; a1 += x * w4.y; a2 += x * w4.z; a3 += x * w4.w;
      }
      float4 r; r.x = a0; r.y = a1; r.z = a2; r.w = a3;
      u4a4[g] = r;
    } else if (tid < 544) {
      int l = tid - 512;
      int seg = bond - last_bond;
      if (seg < 0) seg = 0;
      if (seg > kS - 1) seg = kS - 1;
      float p = 0.f;
      for (int k = l; k < kW; k += 32) p += chunk_emb[(size_t)seg * kW + k] * vt2[k];
      for (int off = 16; off; off >>= 1) p += __shfl_down(p, off, 32);
      if (l == 0) ce_s = p;
    }
    __syncthreads();

    // E) scores[s] = tanh(bsum[s] + u3 + u4)@vt1 + ce for s >= bond
    const float ce = ce_s;
    for (int s = wv; s < kS; s += 32) {
      float sc;
      if (s >= bond) {
        float p = 0.f;
        const float* brow = bsum_b + (size_t)s * kW;
        #pragma unroll
        for (int i = 0; i < 4; ++i) {
          int c = lane + 32 * i;                  // float4 chunk index
          float4 bb = *(const float4*)(brow + 4 * c);
          float4 a3v = u3a4[c];
          float4 a4v = u4a4[c];
          float4 vv  = vt1s4[c];
          p += tanhf(bb.x + a3v.x + a4v.x) * vv.x;
          p += tanhf(bb.y + a3v.y + a4v.y) * vv.y;
          p += tanhf(bb.z + a3v.z + a4v.z) * vv.z;
          p += tanhf(bb.w + a3v.w + a4v.w) * vv.w;
        }
        for (int off = 16; off; off >>= 1) p += __shfl_down(p, off, 32);
        sc = p + ce;
      } else {
        sc = -1e30f;
      }
      if (lane == 0) scores[s] = sc;
    }
    __syncthreads();

    // F) argmax (first occurrence of max, like jnp.argmax)
    if (tid < kS) { rval[tid] = scores[tid]; ridx[tid] = tid; }
    __syncthreads();
    for (int off = kS / 2; off > 0; off >>= 1) {
      if (tid < off) {
        float v1 = rval[tid], v2 = rval[tid + off];
        int   i1 = ridx[tid], i2 = ridx[tid + off];
        if (v2 > v1 || (v2 == v1 && i2 < i1)) { rval[tid] = v2; ridx[tid] = i2; }
      }
      __syncthreads();
    }

    // G) bookkeeping (forced progress; exit once bond passes the sequence end)
    if (tid == 0) {
      int idx = ridx[0];
      int new_bond = (idx > bond) ? idx : bond + 1;
      if (new_bond < kS) outv[new_bond] = 1.f;
      s_last = bond;
      s_bond = new_bond;
      if (new_bond >= kS) s_fin = 1;
    }
    __syncthreads();
    if (s_fin) break;
  }

  // output transposed: out[s, b]
  for (int s = tid; s < kS; s += 1024) out[(size_t)s * kB + b] = outv[s];
}

// ---------------------------------------------------------------------------
extern "C" void kernel_launch(void* const* d_in, const int* in_sizes, int n_in,
                              void* d_out, int out_size, void* d_ws, size_t ws_size,
                              hipStream_t stream) {
  const int*   word_ids = (const int*)d_in[0];
  const float* enc      = (const float*)d_in[1];
  const float* enc_hn   = (const float*)d_in[2];
  const float* emb      = (const float*)d_in[3];
  const float* chunk    = (const float*)d_in[4];
  const float* W1       = (const float*)d_in[5];
  const float* W2       = (const float*)d_in[6];
  const float* W3       = (const float*)d_in[7];
  const float* W4       = (const float*)d_in[8];
  const float* vt1      = (const float*)d_in[9];
  const float* vt2      = (const float*)d_in[10];
  const float* Wx       = (const float*)d_in[11];
  const float* Wh       = (const float*)d_in[12];
  const float* bias     = (const float*)d_in[13];

  float* bsum = (float*)d_ws;            // kB*kS*kW floats = 16 MB scratch
  float* out  = (float*)d_out;           // [S, B] fp32

  // 16384 tiles (512 row-tiles x 32 col-tiles), 8 waves per 256-thread block
  bsum_wmma_kernel<<<2048, 256, 0, stream>>>(enc, word_ids, emb, W1, W2, bsum);

  // one workgroup per batch element, 32 waves each
  decode_kernel<<<kB, 1024, 0, stream>>>(word_ids, enc_hn, emb, chunk,
                                         W3, W4, vt1, vt2, Wx, Wh, bias,
                                         bsum, out);
}